// GNN_node_71794673320190
// MI455X (gfx1250) — compile-verified
//
#include <hip/hip_runtime.h>

typedef float v2f __attribute__((ext_vector_type(2)));
typedef float v8f __attribute__((ext_vector_type(8)));

#define NODES   50000
#define DD      128
#define IN_DIM  64
#define EDGES   640000
#define L_LAYERS 5
#define ROWT    25          // row-tiles per wave pass; 3125 = 25 * 125
#define GRID_Y  125

__device__ __forceinline__ v8f wmma_f32(v2f a, v2f b, v8f c) {
  // D = A(16x4 f32) * B(4x16 f32) + C(16x16 f32)
  return __builtin_amdgcn_wmma_f32_16x16x4_f32(
      /*neg_a=*/false, a, /*neg_b=*/false, b,
      /*c_mod=*/(short)0, c, /*reuse_a=*/false, /*reuse_b=*/false);
}

__device__ __forceinline__ void fadd_atomic(float* p, float v) {
  __hip_atomic_fetch_add(p, v, __ATOMIC_RELAXED, __HIP_MEMORY_SCOPE_AGENT);
}

// ---------------- atom encoder: h = x @ atom_W + atom_b ----------------
__global__ __launch_bounds__(128)
void atom_encoder_kernel(const float* __restrict__ x,
                         const float* __restrict__ W,
                         const float* __restrict__ bias,
                         float* __restrict__ h) {
  const int lane = threadIdx.x & 31;
  const int wave = threadIdx.x >> 5;
  const int half = lane >> 4;
  const int lan  = lane & 15;
  const int col  = blockIdx.x * 64 + wave * 16 + lan;
  const float bv = bias[col];
  for (int rt = 0; rt < ROWT; ++rt) {
    const int row0 = (blockIdx.y * ROWT + rt) * 16;
    const int r = row0 + lan;
    v8f acc;
#pragma unroll
    for (int i = 0; i < 8; ++i) acc[i] = bv;
#pragma unroll
    for (int k = 0; k < IN_DIM; k += 4) {
      const int kk = k + 2 * half;
      const float2 xv = *(const float2*)(x + r * IN_DIM + kk);
      v2f a; a.x = xv.x; a.y = xv.y;
      v2f b; b.x = W[kk * DD + col]; b.y = W[(kk + 1) * DD + col];
      acc = wmma_f32(a, b, acc);
    }
#pragma unroll
    for (int i = 0; i < 8; ++i)
      h[(row0 + i + half * 8) * DD + col] = acc[i];
  }
}

// ---- fused edge kernel: msg = relu(h[src] + edge_attr@bondW + bondb); agg[dst] += msg ----
__global__ __launch_bounds__(256)
void edge_kernel(const float* __restrict__ h,
                 const float* __restrict__ edge_attr,
                 const int* __restrict__ ei,        // [2,E]: src then dst
                 const float* __restrict__ bW,      // [13,128]
                 const float* __restrict__ bb,      // [128]
                 float* __restrict__ agg) {
  const int idx = blockIdx.x * blockDim.x + threadIdx.x;
  const int e = idx >> 5;            // one wave per edge
  if (e >= EDGES) return;
  const int cg = (idx & 31) * 4;     // 4 channels per lane
  const int s = ei[e];
  const int d = ei[EDGES + e];
  float4 acc = *(const float4*)(bb + cg);
#pragma unroll
  for (int j = 0; j < 13; ++j) {
    const float ea = edge_attr[e * 13 + j];
    const float4 w = *(const float4*)(bW + j * DD + cg);
    acc.x = fmaf(ea, w.x, acc.x);
    acc.y = fmaf(ea, w.y, acc.y);
    acc.z = fmaf(ea, w.z, acc.z);
    acc.w = fmaf(ea, w.w, acc.w);
  }
  const float4 hv = *(const float4*)(h + s * DD + cg);
  const float m0 = fmaxf(acc.x + hv.x, 0.0f);
  const float m1 = fmaxf(acc.y + hv.y, 0.0f);
  const float m2 = fmaxf(acc.z + hv.z, 0.0f);
  const float m3 = fmaxf(acc.w + hv.w, 0.0f);
  float* ap = agg + d * DD + cg;
  fadd_atomic(ap + 0, m0);
  fadd_atomic(ap + 1, m1);
  fadd_atomic(ap + 2, m2);
  fadd_atomic(ap + 3, m3);
}

// ---- mlp1: t = ((1+eps)h + agg) @ W1 + b1, accumulate channel sum/sumsq ----
__global__ __launch_bounds__(128)
void mlp1_kernel(const float* __restrict__ h,
                 const float* __restrict__ agg,
                 const float* __restrict__ eps_arr, int layer,
                 const float* __restrict__ W1,      // [128,256]
                 const float* __restrict__ b1,      // [256]
                 float* __restrict__ t,             // [N,256]
                 float* __restrict__ stats) {       // [512]: sum, sumsq
  const float e1 = 1.0f + eps_arr[layer];
  const int lane = threadIdx.x & 31;
  const int wave = threadIdx.x >> 5;
  const int half = lane >> 4;
  const int lan  = lane & 15;
  const int col  = blockIdx.x * 64 + wave * 16 + lan;
  const float bv = b1[col];
  float lsum = 0.0f, lsq = 0.0f;
  for (int rt = 0; rt < ROWT; ++rt) {
    const int row0 = (blockIdx.y * ROWT + rt) * 16;
    const int r = row0 + lan;
    v8f acc;
#pragma unroll
    for (int i = 0; i < 8; ++i) acc[i] = bv;
#pragma unroll 8
    for (int k = 0; k < DD; k += 4) {
      const int kk = k + 2 * half;
      const float2 hv = *(const float2*)(h + r * DD + kk);
      const float2 av = *(const float2*)(agg + r * DD + kk);
      v2f a; a.x = fmaf(e1, hv.x, av.x); a.y = fmaf(e1, hv.y, av.y);
      v2f b; b.x = W1[kk * 256 + col]; b.y = W1[(kk + 1) * 256 + col];
      acc = wmma_f32(a, b, acc);
    }
#pragma unroll
    for (int i = 0; i < 8; ++i) {
      const float v = acc[i];
      t[(row0 + i + half * 8) * 256 + col] = v;
      lsum += v; lsq += v * v;
    }
  }
  fadd_atomic(&stats[col], lsum);
  fadd_atomic(&stats[256 + col], lsq);
}

// ---- bn_finalize: scale = g*rsqrt(var+eps), shift = b - mu*scale ----
__global__ void bn_finalize_kernel(const float* __restrict__ stats,
                                   const float* __restrict__ g,
                                   const float* __restrict__ bb,
                                   float* __restrict__ ss, int C) {
  const int i = threadIdx.x;
  if (i < C) {
    const float invN = 1.0f / (float)NODES;
    const float mu  = stats[i] * invN;
    const float var = stats[C + i] * invN - mu * mu;
    const float rs  = rsqrtf(var + 1e-5f);
    const float sc  = g[i] * rs;
    ss[i]     = sc;
    ss[C + i] = bb[i] - mu * sc;
  }
}

// ---- mlp2: u = relu(BN1(t)) @ W2 + b2, accumulate channel sum/sumsq ----
__global__ __launch_bounds__(128)
void mlp2_kernel(const float* __restrict__ t,      // [N,256]
                 const float* __restrict__ ss1,    // [512]: scale, shift
                 const float* __restrict__ W2,     // [256,128]
                 const float* __restrict__ b2,     // [128]
                 float* __restrict__ u,            // [N,128]
                 float* __restrict__ stats) {      // [256]: sum, sumsq
  const int lane = threadIdx.x & 31;
  const int wave = threadIdx.x >> 5;
  const int half = lane >> 4;
  const int lan  = lane & 15;
  const int col  = blockIdx.x * 64 + wave * 16 + lan;
  const float bv = b2[col];
  float lsum = 0.0f, lsq = 0.0f;
  for (int rt = 0; rt < ROWT; ++rt) {
    const int row0 = (blockIdx.y * ROWT + rt) * 16;
    const int r = row0 + lan;
    v8f acc;
#pragma unroll
    for (int i = 0; i < 8; ++i) acc[i] = bv;
#pragma unroll 8
    for (int k = 0; k < 256; k += 4) {
      const int kk = k + 2 * half;
      const float2 tv = *(const float2*)(t + r * 256 + kk);
      const float2 sc = *(const float2*)(ss1 + kk);
      const float2 sh = *(const float2*)(ss1 + 256 + kk);
      v2f a;
      a.x = fmaxf(fmaf(tv.x, sc.x, sh.x), 0.0f);
      a.y = fmaxf(fmaf(tv.y, sc.y, sh.y), 0.0f);
      v2f b; b.x = W2[kk * DD + col]; b.y = W2[(kk + 1) * DD + col];
      acc = wmma_f32(a, b, acc);
    }
#pragma unroll
    for (int i = 0; i < 8; ++i) {
      const float v = acc[i];
      u[(row0 + i + half * 8) * DD + col] = v;
      lsum += v; lsq += v * v;
    }
  }
  fadd_atomic(&stats[col], lsum);
  fadd_atomic(&stats[128 + col], lsq);
}

// ---- apply BN2 (+relu except last layer), write node features ----
__global__ __launch_bounds__(256)
void apply_out_kernel(const float* __restrict__ u,
                      const float* __restrict__ ss2,   // [256]: scale, shift
                      float* __restrict__ h, int do_relu) {
  const int idx = blockIdx.x * blockDim.x + threadIdx.x;   // N*32 threads
  const int col = (idx & 31) * 4;
  const int off = (idx >> 5) * DD + col;
  const float4 v  = *(const float4*)(u + off);
  const float4 sc = *(const float4*)(ss2 + col);
  const float4 sh = *(const float4*)(ss2 + DD + col);
  float r0 = fmaf(v.x, sc.x, sh.x);
  float r1 = fmaf(v.y, sc.y, sh.y);
  float r2 = fmaf(v.z, sc.z, sh.z);
  float r3 = fmaf(v.w, sc.w, sh.w);
  if (do_relu) {
    r0 = fmaxf(r0, 0.0f); r1 = fmaxf(r1, 0.0f);
    r2 = fmaxf(r2, 0.0f); r3 = fmaxf(r3, 0.0f);
  }
  float4 o; o.x = r0; o.y = r1; o.z = r2; o.w = r3;
  *(float4*)(h + off) = o;
}

extern "C" void kernel_launch(void* const* d_in, const int* in_sizes, int n_in,
                              void* d_out, int out_size, void* d_ws, size_t ws_size,
                              hipStream_t stream) {
  const float* x         = (const float*)d_in[0];
  const int*   ei        = (const int*)  d_in[1];
  const float* edge_attr = (const float*)d_in[2];
  const float* atom_W    = (const float*)d_in[3];
  const float* atom_b    = (const float*)d_in[4];
  const float* bond_W    = (const float*)d_in[5];
  const float* bond_b    = (const float*)d_in[6];
  const float* W1        = (const float*)d_in[7];
  const float* b1        = (const float*)d_in[8];
  const float* bn1_g     = (const float*)d_in[9];
  const float* bn1_b     = (const float*)d_in[10];
  const float* W2        = (const float*)d_in[11];
  const float* b2        = (const float*)d_in[12];
  const float* eps       = (const float*)d_in[13];
  const float* bn_g      = (const float*)d_in[14];
  const float* bn_b      = (const float*)d_in[15];
  float* h = (float*)d_out;                       // live node features [N,128]

  float* agg    = (float*)d_ws;                   // [N,128]
  float* t      = agg + (size_t)NODES * DD;       // [N,256]
  float* u      = t + (size_t)NODES * 256;        // [N,128]
  float* stats1 = u + (size_t)NODES * DD;         // [512]
  float* ss1    = stats1 + 512;                   // [512]
  float* stats2 = ss1 + 512;                      // [256]
  float* ss2    = stats2 + 256;                   // [256]

  const dim3 blk(128);

  atom_encoder_kernel<<<dim3(2, GRID_Y), blk, 0, stream>>>(x, atom_W, atom_b, h);

  for (int l = 0; l < L_LAYERS; ++l) {
    hipMemsetAsync(agg, 0, (size_t)NODES * DD * sizeof(float), stream);
    edge_kernel<<<dim3((EDGES * 32) / 256), dim3(256), 0, stream>>>(
        h, edge_attr, ei, bond_W + (size_t)l * 13 * DD, bond_b + (size_t)l * DD, agg);

    hipMemsetAsync(stats1, 0, 512 * sizeof(float), stream);
    mlp1_kernel<<<dim3(4, GRID_Y), blk, 0, stream>>>(
        h, agg, eps, l, W1 + (size_t)l * DD * 256, b1 + (size_t)l * 256, t, stats1);
    bn_finalize_kernel<<<1, 256, 0, stream>>>(stats1, bn1_g + l * 256, bn1_b + l * 256, ss1, 256);

    hipMemsetAsync(stats2, 0, 256 * sizeof(float), stream);
    mlp2_kernel<<<dim3(2, GRID_Y), blk, 0, stream>>>(
        t, ss1, W2 + (size_t)l * 256 * DD, b2 + (size_t)l * DD, u, stats2);
    bn_finalize_kernel<<<1, 128, 0, stream>>>(stats2, bn_g + l * DD, bn_b + l * DD, ss2, 128);

    apply_out_kernel<<<dim3((NODES * 32) / 256), dim3(256), 0, stream>>>(
        u, ss2, h, (l < L_LAYERS - 1) ? 1 : 0);
  }
}